// comp_loss_48507360641184
// MI455X (gfx1250) — compile-verified
//
#include <hip/hip_runtime.h>
#include <hip/hip_bf16.h>

typedef __attribute__((ext_vector_type(2))) float v2f;
typedef __attribute__((ext_vector_type(8))) float v8f;

#define WIN 11
#define PLANES 48            // 16 batch * 3 channels
#define NSLOT 1024

__device__ __forceinline__ void gauss_taps(float g[WIN]) {
    float s = 0.f;
#pragma unroll
    for (int i = 0; i < WIN; ++i) {
        float d = (float)(i - 5);
        g[i] = expf(-(d * d) / (2.0f * 1.5f * 1.5f));
        s += g[i];
    }
    float inv = 1.0f / s;
#pragma unroll
    for (int i = 0; i < WIN; ++i) g[i] *= inv;
}

// One wave (32 lanes) computes one 16x16 output tile of the valid 11x11
// depthwise Gaussian conv for 5 images (x1, x2, x1^2, x2^2, x1*x2), then the
// SSIM map / CS contributions, reduced into per-level partial-sum slots.
// Vertical conv stage = banded matmul on v_wmma_f32_16x16x4_f32 (7 K-chunks).
__global__ __launch_bounds__(32) void ssim_level_kernel(
    const float* __restrict__ img1, const float* __restrict__ img2,
    int H, int W, int tiles,
    float* __restrict__ partSsim, float* __restrict__ partCs)
{
    const float C1 = 1e-4f;   // 0.01^2
    const float C2 = 9e-4f;   // 0.03^2

    __shared__ float sIn1[26][28];
    __shared__ float sIn2[26][28];
    __shared__ float sH[5][28][16];   // horizontal-conv intermediates, rows 26..27 zero
    __shared__ float sBand[16][28];   // vertical banded weight matrix A

    const int lane = threadIdx.x;
    const int tilesPP = tiles * tiles;
    const int plane = blockIdx.x / tilesPP;
    const int rem   = blockIdx.x - plane * tilesPP;
    const int ty = rem / tiles;
    const int tx = rem - ty * tiles;
    const int outH = H - 10, outW = W - 10;

    float g[WIN];
    gauss_taps(g);

    // ---- band matrix A[m][k] = g[k-m] for 0<=k-m<11, else 0 (rows m=0..15, k=0..27)
    for (int idx = lane; idx < 16 * 28; idx += 32)
        (&sBand[0][0])[idx] = 0.f;
    __syncthreads();
    if (lane < 16) {
#pragma unroll
        for (int d = 0; d < WIN; ++d)
            sBand[lane][lane + d] = g[d];
    }

    // ---- load 26x26 halo tiles (zero-fill OOB; OOB outputs are masked later)
    const long long pbase = (long long)plane * H * W;
    const int r0 = ty * 16, c0 = tx * 16;
    for (int idx = lane; idx < 26 * 26; idx += 32) {
        int r = idx / 26, c = idx - r * 26;
        int gr = r0 + r, gc = c0 + c;
        bool ok = (gr < H) && (gc < W);
        long long off = pbase + (long long)gr * W + gc;
        sIn1[r][c] = ok ? img1[off] : 0.f;
        sIn2[r][c] = ok ? img2[off] : 0.f;
    }
    // zero pad rows 26,27 of the intermediates (band is zero there; avoid NaN*0)
    for (int idx = lane; idx < 5 * 2 * 16; idx += 32) {
        int i = idx / 32, r2 = (idx >> 4) & 1, c = idx & 15;
        sH[i][26 + r2][c] = 0.f;
    }
    __syncthreads();

    // ---- horizontal 11-tap pass: 26 rows x 16 cols, 5 images at once
    for (int idx = lane; idx < 26 * 16; idx += 32) {
        int r = idx >> 4, c = idx & 15;
        float s1 = 0.f, s2 = 0.f, s11 = 0.f, s22 = 0.f, s12 = 0.f;
#pragma unroll
        for (int t = 0; t < WIN; ++t) {
            float a = sIn1[r][c + t];
            float b = sIn2[r][c + t];
            float w = g[t];
            s1  += w * a;
            s2  += w * b;
            s11 += w * a * a;
            s22 += w * b * b;
            s12 += w * a * b;
        }
        sH[0][r][c] = s1;  sH[1][r][c] = s2;
        sH[2][r][c] = s11; sH[3][r][c] = s22; sH[4][r][c] = s12;
    }
    __syncthreads();

    // ---- vertical pass as WMMA: D(16x16) = Band(16x28) x sH(28x16)
    const int half = lane >> 4;      // 0: lanes 0-15, 1: lanes 16-31
    const int idx16 = lane & 15;
    v8f acc[5] = {};
#pragma unroll
    for (int ch = 0; ch < 7; ++ch) {
        const int k0 = 4 * ch + 2 * half;   // A lanes 0-15 hold K{0,1}; 16-31 hold K{2,3}
        v2f a;
        a.x = sBand[idx16][k0];
        a.y = sBand[idx16][k0 + 1];
#pragma unroll
        for (int i = 0; i < 5; ++i) {
            v2f b;                           // B VGPR0: K rows {0,2}; VGPR1: {1,3}
            b.x = sH[i][k0][idx16];
            b.y = sH[i][k0 + 1][idx16];
            acc[i] = __builtin_amdgcn_wmma_f32_16x16x4_f32(
                false, a, false, b, (short)0, acc[i], false, false);
        }
    }

    // ---- epilogue: per-pixel SSIM / CS, masked to the valid output region
    float lS = 0.f, lC = 0.f;
#pragma unroll
    for (int v = 0; v < 8; ++v) {
        int orow = r0 + v + 8 * half;        // D VGPR v: rows v (lanes 0-15) / v+8
        int ocol = c0 + idx16;
        if (orow < outH && ocol < outW) {
            float mu1 = acc[0][v], mu2 = acc[1][v];
            float m11 = mu1 * mu1, m22 = mu2 * mu2, m12 = mu1 * mu2;
            float sg1 = acc[2][v] - m11;
            float sg2 = acc[3][v] - m22;
            float sg12 = acc[4][v] - m12;
            float v1 = 2.0f * sg12 + C2;
            float v2 = sg1 + sg2 + C2;
            lC += v1 / v2;
            lS += ((2.0f * m12 + C1) * v1) / ((m11 + m22 + C1) * v2);
        }
    }
#pragma unroll
    for (int off = 16; off > 0; off >>= 1) {
        lS += __shfl_xor(lS, off, 32);
        lC += __shfl_xor(lC, off, 32);
    }
    if (lane == 0) {
        atomicAdd(&partSsim[blockIdx.x & (NSLOT - 1)], lS);
        atomicAdd(&partCs[blockIdx.x & (NSLOT - 1)], lC);
    }
}

// 2x2 average pool for both images; optionally fuses the MSE partial sums
// (done on the level-0 pass so full-res inputs are read exactly once here).
__global__ void pool_kernel(const float* __restrict__ inA, const float* __restrict__ inB,
                            float* __restrict__ outA, float* __restrict__ outB,
                            int Hin, int Win, float* __restrict__ partMse, int doMse)
{
    const int Ho = Hin >> 1, Wo = Win >> 1;
    const long long nOut = (long long)PLANES * Ho * Wo;
    const long long stride = (long long)gridDim.x * blockDim.x;
    float mseAcc = 0.f;
    for (long long i = (long long)blockIdx.x * blockDim.x + threadIdx.x; i < nOut; i += stride) {
        int x = (int)(i % Wo);
        long long r = i / Wo;
        int y = (int)(r % Ho);
        int p = (int)(r / Ho);
        long long base = ((long long)p * Hin + 2 * y) * Win + 2 * x;
        float a00 = inA[base],       a01 = inA[base + 1];
        float a10 = inA[base + Win], a11 = inA[base + Win + 1];
        float b00 = inB[base],       b01 = inB[base + 1];
        float b10 = inB[base + Win], b11 = inB[base + Win + 1];
        outA[i] = 0.25f * (a00 + a01 + a10 + a11);
        outB[i] = 0.25f * (b00 + b01 + b10 + b11);
        if (doMse) {
            float d0 = a00 - b00, d1 = a01 - b01, d2 = a10 - b10, d3 = a11 - b11;
            mseAcc += d0 * d0 + d1 * d1 + d2 * d2 + d3 * d3;
        }
    }
    if (doMse) {
#pragma unroll
        for (int off = 16; off > 0; off >>= 1)
            mseAcc += __shfl_xor(mseAcc, off, 32);
        if ((threadIdx.x & 31) == 0) {
            int slot = (int)((((long long)blockIdx.x * blockDim.x + threadIdx.x) >> 5) & (NSLOT - 1));
            atomicAdd(&partMse[slot], mseAcc);
        }
    }
}

__global__ void zero_kernel(float* __restrict__ p, int n)
{
    int i = blockIdx.x * blockDim.x + threadIdx.x;
    if (i < n) p[i] = 0.f;
}

__global__ void final_kernel(const float* __restrict__ partSsim,
                             const float* __restrict__ partCs,
                             const float* __restrict__ partMse,
                             float* __restrict__ out)
{
    __shared__ float sbuf[NSLOT];
    __shared__ float res[11];
    for (int a = 0; a < 11; ++a) {
        const float* src = (a < 5) ? (partSsim + a * NSLOT)
                         : (a < 10) ? (partCs + (a - 5) * NSLOT)
                         : partMse;
        sbuf[threadIdx.x] = src[threadIdx.x];
        __syncthreads();
        for (int s = NSLOT / 2; s > 0; s >>= 1) {
            if ((int)threadIdx.x < s) sbuf[threadIdx.x] += sbuf[threadIdx.x + s];
            __syncthreads();
        }
        if (threadIdx.x == 0) res[a] = sbuf[0];
        __syncthreads();
    }
    if (threadIdx.x == 0) {
        const float Wt[5] = {0.0448f, 0.2856f, 0.3001f, 0.2363f, 0.1333f};
        const int Hs[5] = {512, 256, 128, 64, 32};
        float n4 = (float)PLANES * (Hs[4] - 10) * (Hs[4] - 10);
        float mssim4 = res[4] / n4;
        float p2last = powf(mssim4, Wt[4]);
        // literal: prod(pow1[:-1] * pow2[-1])  ->  includes pow2[-1] four times
        float prod = 1.0f;
        for (int l = 0; l < 4; ++l) {
            float n = (float)PLANES * (Hs[l] - 10) * (Hs[l] - 10);
            float mcs = res[5 + l] / n;
            prod *= powf(mcs, Wt[l]) * p2last;
        }
        float mse = res[10] / ((float)PLANES * 512.0f * 512.0f);
        out[0] = mse - prod + 1.0f;   // loss
        out[1] = prod;                // msssim
    }
}

extern "C" void kernel_launch(void* const* d_in, const int* in_sizes, int n_in,
                              void* d_out, int out_size, void* d_ws, size_t ws_size,
                              hipStream_t stream) {
    (void)in_sizes; (void)n_in; (void)out_size; (void)ws_size;
    const float* reconst  = (const float*)d_in[0];
    const float* original = (const float*)d_in[1];
    float* out = (float*)d_out;
    float* ws  = (float*)d_ws;

    float* partSsim = ws;                      // 5 * 1024
    float* partCs   = partSsim + 5 * NSLOT;    // 5 * 1024
    float* partMse  = partCs + 5 * NSLOT;      // 1024
    float* pyr      = partMse + NSLOT;

    const size_t sz1 = (size_t)PLANES * 256 * 256;
    const size_t sz2 = (size_t)PLANES * 128 * 128;
    const size_t sz3 = (size_t)PLANES * 64 * 64;
    const size_t sz4 = (size_t)PLANES * 32 * 32;
    float* l1a = pyr;        float* l1b = l1a + sz1;
    float* l2a = l1b + sz1;  float* l2b = l2a + sz2;
    float* l3a = l2b + sz2;  float* l3b = l3a + sz3;
    float* l4a = l3b + sz3;  float* l4b = l4a + sz4;

    const int nPart = 11 * NSLOT;
    zero_kernel<<<(nPart + 255) / 256, 256, 0, stream>>>(ws, nPart);

    auto ssim = [&](const float* a, const float* b, int H, int lvl) {
        int tiles = ((H - 10) + 15) / 16;
        ssim_level_kernel<<<dim3(tiles * tiles * PLANES), 32, 0, stream>>>(
            a, b, H, H, tiles, partSsim + lvl * NSLOT, partCs + lvl * NSLOT);
    };

    // reference calls _msssim(original, reconst): img1 = original
    ssim(original, reconst, 512, 0);
    pool_kernel<<<2048, 256, 0, stream>>>(original, reconst, l1a, l1b, 512, 512, partMse, 1);
    ssim(l1a, l1b, 256, 1);
    pool_kernel<<<1024, 256, 0, stream>>>(l1a, l1b, l2a, l2b, 256, 256, partMse, 0);
    ssim(l2a, l2b, 128, 2);
    pool_kernel<<<512, 256, 0, stream>>>(l2a, l2b, l3a, l3b, 128, 128, partMse, 0);
    ssim(l3a, l3b, 64, 3);
    pool_kernel<<<256, 256, 0, stream>>>(l3a, l3b, l4a, l4b, 64, 64, partMse, 0);
    ssim(l4a, l4b, 32, 4);

    final_kernel<<<1, NSLOT, 0, stream>>>(partSsim, partCs, partMse, out);
}